// GraphAttentionLayer_29746943492369
// MI455X (gfx1250) — compile-verified
//
#include <hip/hip_runtime.h>
#include <hip/hip_bf16.h>
#include <math.h>

#define D 128

typedef __attribute__((ext_vector_type(2))) float v2f;
typedef __attribute__((ext_vector_type(8))) float v8f;

__device__ __forceinline__ float gelu_tanh(float x) {
    const float c0 = 0.7978845608028654f;   // sqrt(2/pi)
    const float c1 = 0.044715f;
    float x3 = x * x * x;
    return 0.5f * x * (1.0f + tanhf(c0 * (x + c1 * x3)));
}

// -----------------------------------------------------------------------------
// Y[m,n] = gelu( X[m,:] @ W[:,n] + b[n] )   X:[rows,128]  W:[128,128] row-major
// One wave32 owns a 16-row tile and all 128 output columns (8 N-tiles of 16).
// Uses V_WMMA_F32_16X16X4_F32 (full fp32) chained over K=128 in steps of 4.
// VGPR layouts per CDNA5 ISA 7.12.2:
//   A 16x4 f32 : lane(l&15)=M, hi=(l>>4): VGPR0 = A[M][ko+2*hi], VGPR1 = +1
//   B 4x16 f32 : lane(l&15)=N, VGPR j    = B[ko+2*hi+j][N]
//   C/D 16x16  : lane(l&15)=N, VGPR j -> M = j + 8*hi
// rows must be a multiple of 16 (50000 = 3125*16) -> no divergence, EXEC all 1s.
// -----------------------------------------------------------------------------
__global__ void proj_gelu_wmma(const float* __restrict__ X,
                               const float* __restrict__ W,
                               const float* __restrict__ bias,
                               float* __restrict__ Y,
                               int ntiles) {
    const int wave = (blockIdx.x * blockDim.x + threadIdx.x) >> 5;
    if (wave >= ntiles) return;                 // wave-uniform branch
    const int lane = threadIdx.x & 31;
    const int lmod = lane & 15;
    const int hi   = lane >> 4;
    const int row0 = wave * 16;

    v8f acc[8];
    #pragma unroll
    for (int nt = 0; nt < 8; ++nt) acc[nt] = (v8f)(0.0f);

    const float* arow  = X + (size_t)(row0 + lmod) * D;
    const int kshift = 2 * hi;

    for (int ko = 0; ko < D; ko += 4) {
        v2f a;
        const float2 ap = *(const float2*)(arow + ko + kshift);
        a[0] = ap.x; a[1] = ap.y;
        #pragma unroll
        for (int nt = 0; nt < 8; ++nt) {
            const int n = nt * 16 + lmod;
            v2f b;
            b[0] = W[(ko + kshift) * D + n];
            b[1] = W[(ko + kshift + 1) * D + n];
            acc[nt] = __builtin_amdgcn_wmma_f32_16x16x4_f32(
                false, a, false, b, (short)0, acc[nt], false, false);
        }
    }

    #pragma unroll
    for (int nt = 0; nt < 8; ++nt) {
        const int n  = nt * 16 + lmod;
        const float bn = bias[n];
        #pragma unroll
        for (int j = 0; j < 8; ++j) {
            const int m = row0 + j + hi * 8;
            Y[(size_t)m * D + n] = gelu_tanh(acc[nt][j] + bn);
        }
    }
}

// -----------------------------------------------------------------------------
// rows[] is sorted: CSR row offsets via binary search (lower_bound).
// Threads i in [0, N]:  row_ptr[i] = first e with rows[e] >= i; row_ptr[N] = E.
// -----------------------------------------------------------------------------
__global__ void row_ptr_kernel(const int* __restrict__ rows, int E, int N,
                               int* __restrict__ row_ptr) {
    const int i = blockIdx.x * blockDim.x + threadIdx.x;
    if (i > N) return;
    int lo = 0, hiB = E;
    while (lo < hiB) {
        int mid = (lo + hiB) >> 1;
        if (rows[mid] < i) lo = mid + 1; else hiB = mid;
    }
    row_ptr[i] = lo;
}

// -----------------------------------------------------------------------------
// One wave32 per edge: scores[e] = (q[rows[e]] . k[cols[e]]) * adj[e] / sqrt(128)
// Lane l handles 4 channels (float4); tree-reduce with shfl_xor over 32 lanes.
// -----------------------------------------------------------------------------
__global__ void edge_score_kernel(const float* __restrict__ q,
                                  const float* __restrict__ k,
                                  const int* __restrict__ rows,
                                  const int* __restrict__ cols,
                                  const float* __restrict__ adj,
                                  float* __restrict__ scores, int E) {
    const int gid  = blockIdx.x * blockDim.x + threadIdx.x;
    const int e    = gid >> 5;
    const int lane = gid & 31;
    if (e >= E) return;                          // wave-uniform
    const int r = rows[e], c = cols[e];
    const float4 qv = *(const float4*)(q + (size_t)r * D + lane * 4);
    const float4 kv = *(const float4*)(k + (size_t)c * D + lane * 4);
    float d = qv.x * kv.x + qv.y * kv.y + qv.z * kv.z + qv.w * kv.w;
    #pragma unroll
    for (int off = 16; off > 0; off >>= 1) d += __shfl_xor(d, off, 32);
    if (lane == 0) scores[e] = d * adj[e] * 0.08838834764831845f; // 1/sqrt(128)
}

// -----------------------------------------------------------------------------
// One wave32 per row: sparse softmax over the row's edges + weighted sum of v.
// Lane l owns output channels [4l, 4l+3]. Empty rows -> zeros (ref semantics).
// -----------------------------------------------------------------------------
__global__ void row_attend_kernel(const float* __restrict__ scores,
                                  const float* __restrict__ v,
                                  const int* __restrict__ cols,
                                  const int* __restrict__ row_ptr,
                                  float* __restrict__ out, int N) {
    const int w    = (blockIdx.x * blockDim.x + threadIdx.x) >> 5;
    const int lane = threadIdx.x & 31;
    if (w >= N) return;                          // wave-uniform
    const int s0 = row_ptr[w], s1 = row_ptr[w + 1];
    float* orow = out + (size_t)w * D;
    if (s0 == s1) {                              // wave-uniform: no edges
        *(float4*)(orow + lane * 4) = make_float4(0.f, 0.f, 0.f, 0.f);
        return;
    }
    // pass 1: row max
    float m = -3.402823466e38f;
    for (int e = s0 + lane; e < s1; e += 32) m = fmaxf(m, scores[e]);
    #pragma unroll
    for (int off = 16; off > 0; off >>= 1) m = fmaxf(m, __shfl_xor(m, off, 32));
    // pass 2: denom
    float s = 0.f;
    for (int e = s0 + lane; e < s1; e += 32) s += expf(scores[e] - m);
    #pragma unroll
    for (int off = 16; off > 0; off >>= 1) s += __shfl_xor(s, off, 32);
    const float inv = 1.0f / s;
    // pass 3: out_i = sum_e w_e * v[cols[e]]   (v is L2-resident; broadcast w_e)
    float4 acc = make_float4(0.f, 0.f, 0.f, 0.f);
    for (int e = s0; e < s1; ++e) {
        const float wgt = expf(scores[e] - m) * inv;
        const float4 vv = *(const float4*)(v + (size_t)cols[e] * D + lane * 4);
        acc.x += wgt * vv.x; acc.y += wgt * vv.y;
        acc.z += wgt * vv.z; acc.w += wgt * vv.w;
    }
    *(float4*)(orow + lane * 4) = acc;
}

// -----------------------------------------------------------------------------
extern "C" void kernel_launch(void* const* d_in, const int* in_sizes, int n_in,
                              void* d_out, int out_size, void* d_ws, size_t ws_size,
                              hipStream_t stream) {
    const float* query  = (const float*)d_in[0];
    const float* memory = (const float*)d_in[1];
    const float* adj    = (const float*)d_in[2];
    const float* Wq     = (const float*)d_in[3];
    const float* bq     = (const float*)d_in[4];
    const float* Wk     = (const float*)d_in[5];
    const float* bk     = (const float*)d_in[6];
    const float* Wv     = (const float*)d_in[7];
    const float* bv     = (const float*)d_in[8];
    const int*   rows   = (const int*)d_in[9];
    const int*   cols   = (const int*)d_in[10];

    const int N = in_sizes[0] / D;   // 50000
    const int M = in_sizes[1] / D;   // 50000
    const int E = in_sizes[2];       // 1600000

    // workspace layout (all fully written before read)
    float* q      = (float*)d_ws;                       // N*128
    float* k      = q + (size_t)N * D;                  // M*128
    float* v      = k + (size_t)M * D;                  // M*128
    float* scores = v + (size_t)M * D;                  // E
    int*   rptr   = (int*)(scores + E);                 // N+1

    const int ntN = N / 16, ntM = M / 16;               // exact (50000 = 3125*16)

    // projections (fp32 WMMA + bias + gelu)
    proj_gelu_wmma<<<(ntN + 7) / 8, 256, 0, stream>>>(query,  Wq, bq, q, ntN);
    proj_gelu_wmma<<<(ntM + 7) / 8, 256, 0, stream>>>(memory, Wk, bk, k, ntM);
    proj_gelu_wmma<<<(ntM + 7) / 8, 256, 0, stream>>>(memory, Wv, bv, v, ntM);

    // CSR offsets from sorted rows
    row_ptr_kernel<<<(N + 1 + 255) / 256, 256, 0, stream>>>(rows, E, N, rptr);

    // edge scores (one wave per edge)
    {
        long long threads = (long long)E * 32;
        edge_score_kernel<<<(unsigned)((threads + 255) / 256), 256, 0, stream>>>(
            q, k, rows, cols, adj, scores, E);
    }

    // per-row softmax + aggregation (one wave per row)
    {
        long long threads = (long long)N * 32;
        row_attend_kernel<<<(unsigned)((threads + 255) / 256), 256, 0, stream>>>(
            scores, v, cols, rptr, (float*)d_out, N);
    }
}